// BilateralGrid_60851096650105
// MI455X (gfx1250) — compile-verified
//
#include <hip/hip_runtime.h>
#include <stdint.h>

// Problem dims (fixed by the reference)
#define GC 12                 // affine channels
#define GL 8                  // grid depth (luma bins)
#define GH 16                 // grid height
#define GW 16                 // grid width
#define CELLS (GL * GH * GW)  // 2048 cells
#define GELEMS (GC * CELLS)   // 24576 floats = 96 KB (fits one WGP's 320 KB LDS)
#define HI 540
#define WI 960
#define PIX (HI * WI)         // 518400 px per image (divisible by 4)
#define TPB 256               // 8 waves (wave32)
#define VPT 4                 // consecutive pixels per vector group (48 B rgb)
#define GROUPS 4              // groups per thread
#define PPB (TPB * VPT * GROUPS)  // 4096 px per block

// Native clang vector: usable with __builtin_nontemporal_load/store
// (HIP's float4 is a struct and is rejected by the builtin).
typedef float v4f __attribute__((ext_vector_type(4)));

// Raw LDS byte offset for the async-DMA VDST operand (ptrtoint of an
// addrspace(3) pointer yields the workgroup-relative LDS address).
__device__ __forceinline__ uint32_t lds_u32(const void* p) {
  return (uint32_t)(uintptr_t)(__attribute__((address_space(3))) const void*)p;
}

__global__ __launch_bounds__(TPB) void bgrid_slice_kernel(
    const float* __restrict__ grids,    // (N, 12, L, H, W)
    const float* __restrict__ grid_xy,  // (N, Hi, Wi, 2)
    const float* __restrict__ rgb,      // (N, Hi, Wi, 3)
    const int* __restrict__ idx,        // (N,)
    float* __restrict__ out)            // (N, Hi, Wi, 3)
{
  // Channel-last in LDS: sg[cell*12 + c]; each cell's 12 floats = 48 B,
  // so v4f reads below are 16 B aligned.
  __shared__ float sg[GELEMS];

  const int n = blockIdx.y;
  const int gsel = idx[n];
  const uint64_t gaddr = (uint64_t)(uintptr_t)(grids + (size_t)gsel * GELEMS);
  const uint32_t lds_base = lds_u32(&sg[0]);

  // ---- Stage one image's grid into LDS via async global->LDS DMA ----
  // Global side: linear element index e (perfectly coalesced b32 per lane).
  // LDS side: transpose (c, cell) -> (cell, c) scatter, done by the DMA.
  #pragma unroll 4
  for (int i = 0; i < GELEMS / TPB; ++i) {
    const int e = i * TPB + (int)threadIdx.x;
    const int c = e >> 11;            // e / 2048
    const int cell = e & (CELLS - 1); // e % 2048
    const uint32_t ldsa = lds_base + (uint32_t)(cell * GC + c) * 4u;
    const uint32_t goff = (uint32_t)e * 4u;
    // GVS mode: mem = SADDR(64) + VADDR(32) ; VDST = LDS byte address.
    asm volatile("global_load_async_to_lds_b32 %0, %1, %2"
                 :: "v"(ldsa), "v"(goff), "s"(gaddr)
                 : "memory");
  }
  asm volatile("s_wait_asynccnt 0x0" ::: "memory"); // drain this wave's DMA
  __syncthreads();                                   // make LDS visible WG-wide

  // ---- Stream pixels: 4 consecutive pixels per lane -> pure b128 traffic ----
  const size_t imgoff = (size_t)n * (size_t)PIX;
  const float* __restrict__ gxyp = grid_xy + imgoff * 2;
  const float* __restrict__ rgbp = rgb + imgoff * 3;
  float* __restrict__ outp = out + imgoff * 3;
  const int tileBase = (int)blockIdx.x * PPB;

  #pragma unroll
  for (int gI = 0; gI < GROUPS; ++gI) {
    const int p0 = tileBase + (gI * TPB + (int)threadIdx.x) * VPT;
    if (p0 >= PIX) break;  // PIX % 4 == 0 -> groups are fully in or out

    // Coalesced, read-once streams: lane t covers bytes [48t, 48t+48).
    const v4f* rp = (const v4f*)(rgbp + 3 * p0);
    const v4f r0 = __builtin_nontemporal_load(rp + 0);
    const v4f r1 = __builtin_nontemporal_load(rp + 1);
    const v4f r2 = __builtin_nontemporal_load(rp + 2);
    const v4f* xp = (const v4f*)(gxyp + 2 * p0);
    const v4f x01 = __builtin_nontemporal_load(xp + 0);  // x0 y0 x1 y1
    const v4f x23 = __builtin_nontemporal_load(xp + 1);  // x2 y2 x3 y3

    const float pr[12] = {r0.x, r0.y, r0.z, r0.w, r1.x, r1.y,
                          r1.z, r1.w, r2.x, r2.y, r2.z, r2.w};
    const float px[4] = {x01.x, x01.z, x23.x, x23.z};
    const float py[4] = {x01.y, x01.w, x23.y, x23.w};
    float of[12];

    #pragma unroll
    for (int j = 0; j < VPT; ++j) {
      const float r = pr[3 * j + 0];
      const float g = pr[3 * j + 1];
      const float b = pr[3 * j + 2];
      const float gray = fmaf(0.299f, r, fmaf(0.587f, g, 0.114f * b));

      // align_corners=True mapping with border clamp (matches reference)
      const float x = fminf(fmaxf(px[j] * (float)(GW - 1), 0.0f), (float)(GW - 1));
      const float y = fminf(fmaxf(py[j] * (float)(GH - 1), 0.0f), (float)(GH - 1));
      const float z = fminf(fmaxf(gray * (float)(GL - 1), 0.0f), (float)(GL - 1));
      const int x0 = min((int)x, GW - 2);
      const int y0 = min((int)y, GH - 2);
      const int z0 = min((int)z, GL - 2);
      const float fx = x - (float)x0;
      const float fy = y - (float)y0;
      const float fz = z - (float)z0;

      float a[12];
      #pragma unroll
      for (int q = 0; q < 12; ++q) a[q] = 0.0f;

      const int base = ((z0 * GH + y0) * GW + x0) * GC;
      #pragma unroll
      for (int cz = 0; cz < 2; ++cz) {
        const float wz = cz ? fz : 1.0f - fz;
        #pragma unroll
        for (int cy = 0; cy < 2; ++cy) {
          const float wyz = wz * (cy ? fy : 1.0f - fy);
          #pragma unroll
          for (int cx = 0; cx < 2; ++cx) {
            const float w = wyz * (cx ? fx : 1.0f - fx);
            const v4f* cp = (const v4f*)&sg[base + cz * (GH * GW * GC) +
                                           cy * (GW * GC) + cx * GC];
            const v4f v0 = cp[0];  // 3x ds_load_b128 per corner
            const v4f v1 = cp[1];
            const v4f v2 = cp[2];
            a[0]  = fmaf(w, v0.x, a[0]);  a[1]  = fmaf(w, v0.y, a[1]);
            a[2]  = fmaf(w, v0.z, a[2]);  a[3]  = fmaf(w, v0.w, a[3]);
            a[4]  = fmaf(w, v1.x, a[4]);  a[5]  = fmaf(w, v1.y, a[5]);
            a[6]  = fmaf(w, v1.z, a[6]);  a[7]  = fmaf(w, v1.w, a[7]);
            a[8]  = fmaf(w, v2.x, a[8]);  a[9]  = fmaf(w, v2.y, a[9]);
            a[10] = fmaf(w, v2.z, a[10]); a[11] = fmaf(w, v2.w, a[11]);
          }
        }
      }

      // Per-pixel 3x4 affine: out = A[:, :3] @ rgb + A[:, 3]
      of[3 * j + 0] = fmaf(a[0], r, fmaf(a[1], g, fmaf(a[2],  b, a[3])));
      of[3 * j + 1] = fmaf(a[4], r, fmaf(a[5], g, fmaf(a[6],  b, a[7])));
      of[3 * j + 2] = fmaf(a[8], r, fmaf(a[9], g, fmaf(a[10], b, a[11])));
    }

    v4f* op = (v4f*)(outp + 3 * p0);
    const v4f w0 = {of[0], of[1], of[2],  of[3]};
    const v4f w1 = {of[4], of[5], of[6],  of[7]};
    const v4f w2 = {of[8], of[9], of[10], of[11]};
    __builtin_nontemporal_store(w0, op + 0);
    __builtin_nontemporal_store(w1, op + 1);
    __builtin_nontemporal_store(w2, op + 2);
  }
}

extern "C" void kernel_launch(void* const* d_in, const int* in_sizes, int n_in,
                              void* d_out, int out_size, void* d_ws, size_t ws_size,
                              hipStream_t stream) {
  const float* grids   = (const float*)d_in[0];
  const float* grid_xy = (const float*)d_in[1];
  const float* rgb     = (const float*)d_in[2];
  const int*   idx     = (const int*)d_in[3];
  float* out = (float*)d_out;

  const int N = in_sizes[3];  // number of views
  dim3 grid((PIX + PPB - 1) / PPB, N, 1);
  bgrid_slice_kernel<<<grid, dim3(TPB, 1, 1), 0, stream>>>(grids, grid_xy, rgb,
                                                           idx, out);
}